// PScanTriton_45028437131366
// MI455X (gfx1250) — compile-verified
//
#include <hip/hip_runtime.h>
#include <stdint.h>

// ---------------------------------------------------------------------------
// Complex linear recurrence  y_t = A_t y_{t-1} + x_t   (B=64, L=512, D=32)
//
// Memory-bound: 256 MB of A streamed once (~12 us at 23.3 TB/s floor).
// One workgroup per batch, ONE wave per workgroup. The Tensor Data Mover
// (tensor_load_to_lds) prefetches A chunks into a deep LDS ring (8 slots x
// 4 steps, ~295 KB -> exactly one WG per WGP), synchronized with
// s_wait_tensorcnt. TDM's pad feature inserts 4 DWORDs every 32 DWORDs so
// the LDS row stride is 144 B => conflict-minimal b128 row reads.
// The sequential matvec runs on v_pk_fma_f32 (inline asm; generic <2 x float>
// fma scalarized), seeded by v_pk_mul_f32, with 4 independent accumulator
// chains so a lone wave never stalls on pk-FMA latency.
// f32 precision is required by the 512-step chain; WMMA's only f32 shape,
// 16x16x4, would cost >= as many issue slots as pk-FMA while wasting 15/16
// of the N dimension -> TDM is the CDNA5 feature that pays here, not WMMA.
// ---------------------------------------------------------------------------

typedef __attribute__((ext_vector_type(2))) float        f32x2;
typedef __attribute__((ext_vector_type(4))) float        f32x4;
typedef __attribute__((ext_vector_type(4))) unsigned int v4u;
typedef __attribute__((ext_vector_type(8))) int          v8i;
typedef __attribute__((ext_vector_type(4))) int          v4i;

#define BB 64
#define LL 512
#define DD 32
#define SS 4               // time steps per TDM chunk
#define CC (LL / SS)       // 128 chunks per batch
#define RR 8               // LDS ring slots

#define ROWB  144u                    // 32 DW data + 4 DW pad = 36 DW
#define STEPB (32u * ROWB)            // 4608 B  (one 32x32 f32 matrix, padded)
#define ARRB  ((unsigned)SS * STEPB)  // 18432 B (one chunk of one array)
#define SLOTB (2u * ARRB)             // 36864 B (A_re chunk + A_im chunk)
#define RINGB ((unsigned)RR * SLOTB)  // 294912 B

// D = S0*S1 + D  (packed 2xf32)
#define PK_FMA(acc, a, b) \
  asm("v_pk_fma_f32 %0, %1, %2, %0" : "+v"(acc) : "v"(a), "v"(b))
// D = (-S0)*S1 + D  (negate both halves of src0 via VOP3P neg modifiers)
#define PK_FMA_NEG(acc, a, b) \
  asm("v_pk_fma_f32 %0, %1, %2, %0 neg_lo:[1,0,0] neg_hi:[1,0,0]" \
      : "+v"(acc) : "v"(a), "v"(b))
// D = S0*S1  (chain seed: avoids zero-init movs)
#define PK_MUL(acc, a, b) \
  asm("v_pk_mul_f32 %0, %1, %2" : "=v"(acc) : "v"(a), "v"(b))

// One TDM load: contiguous 16 KB (4096 f32) from global -> LDS, padded
// 4 DWORDs after every 32 DWORDs (pad_interval=4 -> 32 DW, pad_amount=3 -> 4 DW).
static __device__ __forceinline__ void tdm_load_chunk(unsigned long long gaddr,
                                                      unsigned lds_addr)
{
  v4u g0;
  g0.x = 1u;                                        // count=1 (valid), user mode
  g0.y = lds_addr;                                  // LDS byte address
  g0.z = (unsigned)(gaddr & 0xFFFFFFFFull);         // global_addr[31:0]
  g0.w = (unsigned)((gaddr >> 32) & 0x1FFFFFFull)   // global_addr[56:32]
       | 0x80000000u;                               // type = 2 ("image")
  v8i g1;
  g1[0] = 0x07120000;  // pad_amount=3, pad_interval=4, pad_en=1, data_size=2 (4B)
  g1[1] = 0x10000000;  // tensor_dim0 = 4096 (bits[31:16] of this dword)
  g1[2] = 0x00010000;  // tensor_dim1 = 1
  g1[3] = 0x10000000;  // tile_dim0   = 4096 (bits[31:16])
  g1[4] = 0;           // tile_dim1 = tile_dim2 = 0 (1-D tile)
  g1[5] = 0x00001000;  // tensor_dim0_stride = 4096
  g1[6] = 0;
  g1[7] = 0;
  v4i z4;  z4[0] = 0; z4[1] = 0; z4[2] = 0; z4[3] = 0;   // groups 2/3 unused
  v8i z8;  z8[0] = 0; z8[1] = 0; z8[2] = 0; z8[3] = 0;   // extra group (6-arg
  z8[4] = 0; z8[5] = 0; z8[6] = 0; z8[7] = 0;            // toolchain variant)
  __builtin_amdgcn_tensor_load_to_lds(g0, g1, z4, z4, z8, /*cpol*/0);
}

__global__ __launch_bounds__(32) void
pscan_tdm_kernel(const float* __restrict__ Are, const float* __restrict__ Aim,
                 const float* __restrict__ Xre, const float* __restrict__ Xim,
                 float* __restrict__ Y)
{
  __shared__ __align__(16) char smem[RINGB + 256];   // ring + y broadcast buf
  const int lane = threadIdx.x;                      // 0..31 == row index i
  const int b    = blockIdx.x;                       // batch

  float* yrb = (float*)(smem + RINGB);               // y_{t-1}.re [32]
  float* yib = yrb + 32;                             // y_{t-1}.im [32]
  yrb[lane] = 0.0f;                                  // y_{-1} = 0
  yib[lane] = 0.0f;

  // Generic->flat pointer of LDS has the byte offset in the low 32 bits.
  const unsigned lds0 = (unsigned)(unsigned long long)(uintptr_t)smem;

  const unsigned long long aBytes = (unsigned long long)b * (LL * DD * DD) * 4ull;
  const unsigned long long arBase = (unsigned long long)(uintptr_t)Are + aBytes;
  const unsigned long long aiBase = (unsigned long long)(uintptr_t)Aim + aBytes;

  // Prime the TDM pipeline: fill all RR ring slots (2 tensor ops per chunk).
  for (int c0 = 0; c0 < RR; ++c0) {
    const unsigned slot = lds0 + (unsigned)c0 * SLOTB;
    const unsigned long long gofs = (unsigned long long)c0 * (SS * DD * DD * 4);
    tdm_load_chunk(arBase + gofs, slot);
    tdm_load_chunk(aiBase + gofs, slot + ARRB);
  }

  // Double-buffered x prefetch (tiny: 256 B per step per batch).
  float xr[SS], xi[SS];
  {
    const int base = b * (LL * DD) + lane;
#pragma unroll
    for (int s = 0; s < SS; ++s) {
      xr[s] = Xre[base + s * DD];
      xi[s] = Xim[base + s * DD];
    }
  }

  for (int c = 0; c < CC; ++c) {
    float xrn[SS] = {}, xin[SS] = {};
    if (c + 1 < CC) {
      const int base = b * (LL * DD) + (c + 1) * SS * DD + lane;
#pragma unroll
      for (int s = 0; s < SS; ++s) {
        xrn[s] = Xre[base + s * DD];
        xin[s] = Xim[base + s * DD];
      }
    }

    // Chunk c ready when at most 2*(RR-1) tensor ops are still outstanding
    // (tensor ops complete in order; 2 ops per chunk).
    if (c + RR < CC) __builtin_amdgcn_s_wait_tensorcnt(2 * (RR - 1));
    else             __builtin_amdgcn_s_wait_tensorcnt(0);
    __asm__ volatile("" ::: "memory");

    const char* slot = smem + (unsigned)(c & (RR - 1)) * SLOTB;

#pragma unroll
    for (int s = 0; s < SS; ++s) {
      // Lane i reads its own padded row (stride 144 B, 16B-aligned b128 loads).
      const f32x4* arp  = (const f32x4*)(slot + (unsigned)s * STEPB + (unsigned)lane * ROWB);
      const f32x4* aip  = (const f32x4*)(slot + ARRB + (unsigned)s * STEPB + (unsigned)lane * ROWB);
      const f32x4* yr4p = (const f32x4*)yrb;   // broadcast reads (uniform addr)
      const f32x4* yi4p = (const f32x4*)yib;

      // 4 independent accumulator chains (even/odd quartet x re/im): a single
      // wave has no other wave to hide VALU latency behind, so keep >=4-deep
      // independence between consecutive same-chain pk-FMAs. Chains are
      // seeded with v_pk_mul_f32 (no zero-init movs).
      f32x2 accr0, accr1, acci0, acci1;

      auto quartet = [&](int q, f32x2& ar_acc, f32x2& ai_acc, bool seed) {
        f32x4 a4 = arp[q];                     // Ar[i][4q..4q+3]
        f32x4 b4 = aip[q];                     // Ai[i][4q..4q+3]
        f32x4 r4 = yr4p[q];                    // y.re[4q..4q+3] (broadcast)
        f32x4 i4 = yi4p[q];                    // y.im[4q..4q+3] (broadcast)
        f32x2 alo = {a4.x, a4.y}, ahi = {a4.z, a4.w};
        f32x2 blo = {b4.x, b4.y}, bhi = {b4.z, b4.w};
        f32x2 rlo = {r4.x, r4.y}, rhi = {r4.z, r4.w};
        f32x2 ilo = {i4.x, i4.y}, ihi = {i4.z, i4.w};
        if (seed) {
          PK_MUL(ar_acc, alo, rlo);
          PK_MUL(ai_acc, alo, ilo);
        } else {
          PK_FMA(ar_acc, alo, rlo);
          PK_FMA(ai_acc, alo, ilo);
        }
        PK_FMA(ar_acc, ahi, rhi);
        PK_FMA(ai_acc, ahi, ihi);
        PK_FMA_NEG(ar_acc, blo, ilo);
        PK_FMA(ai_acc, blo, rlo);
        PK_FMA_NEG(ar_acc, bhi, ihi);
        PK_FMA(ai_acc, bhi, rhi);
      };

      quartet(0, accr0, acci0, true);
      quartet(1, accr1, acci1, true);
#pragma unroll
      for (int q = 2; q < 8; ++q)
        quartet(q, (q & 1) ? accr1 : accr0, (q & 1) ? acci1 : acci0, false);

      const f32x2 accr = accr0 + accr1;
      const f32x2 acci = acci0 + acci1;
      const float nyr = accr.x + accr.y + xr[s];
      const float nyi = acci.x + acci.y + xi[s];

      // Publish y_t for next step (same-wave LDS is in-order: no barrier).
      yrb[lane] = nyr;
      yib[lane] = nyi;

      const int t = c * SS + s;
      f32x2 o = {nyr, nyi};                    // complex64 interleaved
      *(f32x2*)(Y + ((size_t)(b * LL + t) * DD + lane) * 2) = o;
    }

    // Recycle this slot: our ds reads must drain before TDM overwrites it.
    if (c + RR < CC) {
      __asm__ volatile("s_wait_dscnt 0" ::: "memory");
      const unsigned slotl = lds0 + (unsigned)((c + RR) & (RR - 1)) * SLOTB;
      const unsigned long long gofs =
          (unsigned long long)(c + RR) * (SS * DD * DD * 4);
      tdm_load_chunk(arBase + gofs, slotl);
      tdm_load_chunk(aiBase + gofs, slotl + ARRB);
    }

#pragma unroll
    for (int s = 0; s < SS; ++s) { xr[s] = xrn[s]; xi[s] = xin[s]; }
  }
}

extern "C" void kernel_launch(void* const* d_in, const int* in_sizes, int n_in,
                              void* d_out, int out_size, void* d_ws, size_t ws_size,
                              hipStream_t stream)
{
  (void)in_sizes; (void)n_in; (void)out_size; (void)d_ws; (void)ws_size;
  const float* Are = (const float*)d_in[0];
  const float* Aim = (const float*)d_in[1];
  const float* Xre = (const float*)d_in[2];
  const float* Xim = (const float*)d_in[3];
  float* Y = (float*)d_out;
  pscan_tdm_kernel<<<dim3(BB), dim3(32), 0, stream>>>(Are, Aim, Xre, Xim, Y);
}